// QNet_3564822855711
// MI455X (gfx1250) — compile-verified
//
#include <hip/hip_runtime.h>
#include <hip/hip_bf16.h>
#include <math.h>

typedef __attribute__((ext_vector_type(2))) float v2f;
typedef __attribute__((ext_vector_type(8))) float v8f;

#define BATCH   4096
#define SEQ_T   50
#define MTOT    (BATCH * SEQ_T)      // 204800 rows
#define HPAD    16                   // H=10 padded to 16
#define NQ      10
#define DIMQ    1024
#define TILES   4                    // 16-row tiles per wave
#define WAVES   8                    // waves per block
#define ROWS_PER_BLOCK (16 * TILES * WAVES)   // 512

// ---------------------------------------------------------------------------
// Kernel 1: U[row, 0..15] = X[row, :] @ W_ih^T + (b_ih + b_hh), via f32 WMMA.
// K is compile-time: main loop has NO bounds checks (straight-line loads),
// single compile-time tail step uses clamp+select (v_cndmask, no branches).
// W staged zero-padded in LDS: B-fragment loads are unconditional ds_loads.
// Each wave owns 4 independent 16-row tiles -> 4 WMMAs per B fragment.
// ---------------------------------------------------------------------------
template <int K>
__global__ __launch_bounds__(256) void rnn_in_gemm(
    const float* __restrict__ X,      // [MTOT, K]
    const float* __restrict__ W,      // [10, K]
    const float* __restrict__ b_ih,   // [10]
    const float* __restrict__ b_hh,   // [10]
    float* __restrict__ U)            // [MTOT, 16]
{
  constexpr int KMAIN = K & ~3;
  constexpr int KP    = KMAIN + ((K & 3) ? 4 : 0);   // K rounded up to mult of 4

  __shared__ float wlds[16 * KP];                    // [n][k], zero padded

  const int tid = threadIdx.x;
  // zero-fill (covers rows 10..15 and k >= K padding)
  for (int i = tid; i < 16 * KP; i += 256) wlds[i] = 0.0f;
  __syncthreads();
  // stage W[10][K] with row stride KP
  for (int i = tid; i < NQ * K; i += 256)
    wlds[(i / K) * KP + (i % K)] = W[i];
  __syncthreads();

  const int lane = tid & 31;
  const int wave = tid >> 5;
  const int n    = lane & 15;     // A: row-in-tile; B/D: output column
  const int kh   = lane >> 4;     // K-pair selector (0 or 1)

  const int rowBase = blockIdx.x * ROWS_PER_BLOCK + wave * (16 * TILES);

  const float* xrow[TILES];
#pragma unroll
  for (int t = 0; t < TILES; ++t)
    xrow[t] = X + (size_t)(rowBase + t * 16 + n) * K;

  v8f acc[TILES];
#pragma unroll
  for (int t = 0; t < TILES; ++t) acc[t] = (v8f){};

  const float* wl = wlds + n * KP;

  for (int k0 = 0; k0 < KMAIN; k0 += 4) {
    const int ka = k0 + kh * 2;
    v2f b;
    b.x = wl[ka];
    b.y = wl[ka + 1];
#pragma unroll
    for (int t = 0; t < TILES; ++t) {
      v2f a;
      a.x = xrow[t][ka];
      a.y = xrow[t][ka + 1];
      acc[t] = __builtin_amdgcn_wmma_f32_16x16x4_f32(
          false, a, false, b, (short)0, acc[t], false, false);
    }
  }

  if constexpr (K & 3) {          // single tail step, branch-free loads
    const int ka  = KMAIN + kh * 2;
    const int i0  = (ka     < K) ? ka     : 0;
    const int i1  = (ka + 1 < K) ? ka + 1 : 0;
    v2f b;
    b.x = wl[ka];                 // LDS is zero-padded: no guard needed
    b.y = wl[ka + 1];
#pragma unroll
    for (int t = 0; t < TILES; ++t) {
      float ax = xrow[t][i0];
      float ay = xrow[t][i1];
      v2f a;
      a.x = (ka     < K) ? ax : 0.0f;
      a.y = (ka + 1 < K) ? ay : 0.0f;
      acc[t] = __builtin_amdgcn_wmma_f32_16x16x4_f32(
          false, a, false, b, (short)0, acc[t], false, false);
    }
  }

  // bias: clamp index to stay in-bounds, select 0 for pad columns
  const int  nc   = (n < NQ) ? n : 0;
  const float bsum = b_ih[nc] + b_hh[nc];
  const float bias = (n < NQ) ? bsum : 0.0f;

#pragma unroll
  for (int t = 0; t < TILES; ++t) {
    float* outp = U + (size_t)(rowBase + t * 16) * HPAD + n;
#pragma unroll
    for (int v = 0; v < 8; ++v) {
      const int m = v + 8 * kh;
      outp[(size_t)m * HPAD] = acc[t][v] + bias;
    }
  }
}

// ---------------------------------------------------------------------------
// Kernel 2: sequential tanh-Elman recurrence; one thread per sample.
// h_t = tanh(U[t] + h_{t-1} @ W_hh^T)   (biases pre-folded into U)
// ---------------------------------------------------------------------------
__global__ __launch_bounds__(256) void rnn_recur(
    const float* __restrict__ U,      // [BATCH*SEQ_T, 16]
    const float* __restrict__ W_hh,   // [10,10]
    float* __restrict__ x_p,          // [BATCH, 30]
    int col_off)
{
  __shared__ float w[NQ][NQ];
  if (threadIdx.x < NQ * NQ)
    w[threadIdx.x / NQ][threadIdx.x % NQ] = W_hh[threadIdx.x];
  __syncthreads();

  const int s = blockIdx.x * blockDim.x + threadIdx.x;
  float h[NQ];
#pragma unroll
  for (int j = 0; j < NQ; ++j) h[j] = 0.0f;

  const float* u = U + (size_t)s * SEQ_T * HPAD;
  for (int t = 0; t < SEQ_T; ++t) {
    float nh[NQ];
#pragma unroll
    for (int j = 0; j < NQ; ++j) {
      float a = u[j];
#pragma unroll
      for (int i = 0; i < NQ; ++i) a += h[i] * w[j][i];
      nh[j] = tanhf(a);
    }
#pragma unroll
    for (int j = 0; j < NQ; ++j) h[j] = nh[j];
    u += HPAD;
  }

  const float halfpi = 1.57079632679489662f;
#pragma unroll
  for (int j = 0; j < NQ; ++j)
    x_p[(size_t)s * 30 + col_off + j] = (h[j] + 1.0f) * halfpi;
}

// ---------------------------------------------------------------------------
// Kernel 3: 10-qubit statevector VQC, one workgroup (512 thr) per sample.
// State (1024 complex) in LDS; barrier between gates. Circuit structure
// hardcoded from _make_design(). Readout folded to a single scalar.
// ---------------------------------------------------------------------------
__global__ __launch_bounds__(512) void vqc(
    const float* __restrict__ x_p,    // [BATCH, 30]  ang[q][c] = x_p[3q+c]
    const float* __restrict__ qp,     // [20]
    const float* __restrict__ W_r,    // [10]
    const float* __restrict__ b_r,    // [1]
    float* __restrict__ out)          // [BATCH]
{
  __shared__ float sre[DIMQ];
  __shared__ float sim[DIMQ];
  __shared__ float red[16];

  const int tid = threadIdx.x;
  const int s   = blockIdx.x;
  const float* ang = x_p + (size_t)s * 30;

  sre[tid] = 0.0f; sre[tid + 512] = 0.0f;
  sim[tid] = 0.0f; sim[tid + 512] = 0.0f;
  if (tid == 0) sre[0] = 1.0f;

  // generic 2x2 complex gate on wire w (wire 0 = MSB)
  auto apply2 = [&](int w, float u00r, float u00i, float u01r, float u01i,
                    float u10r, float u10i, float u11r, float u11i) {
    __syncthreads();
    const int bp = 9 - w;
    const int R  = 1 << bp;
    const int i0 = ((tid >> bp) << (bp + 1)) | (tid & (R - 1));
    const int i1 = i0 | R;
    const float a0r = sre[i0], a0i = sim[i0];
    const float a1r = sre[i1], a1i = sim[i1];
    sre[i0] = u00r * a0r - u00i * a0i + u01r * a1r - u01i * a1i;
    sim[i0] = u00r * a0i + u00i * a0r + u01r * a1i + u01i * a1r;
    sre[i1] = u10r * a0r - u10i * a0i + u11r * a1r - u11i * a1i;
    sim[i1] = u10r * a0i + u10i * a0r + u11r * a1i + u11i * a1r;
  };

  // qml.Rot(phi, theta, omega) = RZ(omega) RY(theta) RZ(phi)
  auto rot = [&](int w, float phi, float th, float om) {
    const float c  = cosf(th * 0.5f), sn = sinf(th * 0.5f);
    const float a  = 0.5f * (phi + om), b = 0.5f * (phi - om);
    const float ca = cosf(a), sa = sinf(a);
    const float cb = cosf(b), sb = sinf(b);
    //  ep = ca - i*sa ;  em = cb + i*sb
    apply2(w, ca * c, -sa * c,          // U00 =  ep * c
              -cb * sn, -sb * sn,       // U01 = -em * s
              cb * sn, -sb * sn,        // U10 =  conj(em) * s
              ca * c,  sa * c);         // U11 =  conj(ep) * c
  };

  auto rx = [&](int w, float th) {
    const float c = cosf(th * 0.5f), sn = sinf(th * 0.5f);
    apply2(w, c, 0.f, 0.f, -sn, 0.f, -sn, c, 0.f);
  };
  auto ry = [&](int w, float th) {
    const float c = cosf(th * 0.5f), sn = sinf(th * 0.5f);
    apply2(w, c, 0.f, -sn, 0.f, sn, 0.f, c, 0.f);
  };
  // diagonal IsingZZ(th) on wires (p,q): amp *= exp(-i*th/2 * zp*zq)
  auto zz = [&](int p, int q, float th) {
    __syncthreads();
    const float c = cosf(th * 0.5f), sn = sinf(th * 0.5f);
#pragma unroll
    for (int k = 0; k < 2; ++k) {
      const int i = tid + k * 512;
      const float z = (((i >> (9 - p)) ^ (i >> (9 - q))) & 1) ? -1.0f : 1.0f;
      const float re = sre[i], im = sim[i];
      sre[i] = re * c + im * z * sn;
      sim[i] = im * c - re * z * sn;
    }
  };
  auto rotD = [&](int w) { rot(w, ang[3 * w], ang[3 * w + 1], ang[3 * w + 2]); };

  // ---- initial per-qubit data Rot layer ----
#pragma unroll
  for (int q = 0; q < NQ; ++q) rotD(q);

  // ---- 20 parameterized layers (static design from _make_design) ----
  zz(0, 1, qp[0]);   rx(1, qp[1]);
  zz(1, 2, qp[2]);   ry(3, qp[3]);
  zz(2, 3, qp[4]);   rotD(5);
  zz(3, 4, qp[6]);   rx(7, qp[7]);
  zz(4, 5, qp[8]);   ry(9, qp[9]);
  zz(5, 6, qp[10]);  rotD(1);
  zz(6, 7, qp[12]);  rx(3, qp[13]);
  zz(7, 8, qp[14]);  ry(5, qp[15]);
  zz(8, 9, qp[16]);  rotD(7);
  zz(9, 0, qp[18]);  rx(9, qp[19]);

  // ---- readout: sum_i |amp_i|^2 * (sum_q W_r[q] * z(i,q)) ----
  __syncthreads();
  float wr[NQ];
#pragma unroll
  for (int q = 0; q < NQ; ++q) wr[q] = W_r[q];

  float acc = 0.0f;
#pragma unroll
  for (int k = 0; k < 2; ++k) {
    const int i = tid + k * 512;
    const float p = sre[i] * sre[i] + sim[i] * sim[i];
    float coef = 0.0f;
#pragma unroll
    for (int q = 0; q < NQ; ++q)
      coef += ((i >> (9 - q)) & 1) ? -wr[q] : wr[q];
    acc += p * coef;
  }
  // deterministic wave32 reduction, then serial combine of 16 waves
#pragma unroll
  for (int off = 16; off > 0; off >>= 1)
    acc += __shfl_down(acc, off, 32);
  if ((tid & 31) == 0) red[tid >> 5] = acc;
  __syncthreads();
  if (tid == 0) {
    float v = 0.0f;
#pragma unroll
    for (int i = 0; i < 16; ++i) v += red[i];
    out[s] = tanhf(v + b_r[0]) * 3.0f;
  }
}

// ---------------------------------------------------------------------------
extern "C" void kernel_launch(void* const* d_in, const int* in_sizes, int n_in,
                              void* d_out, int out_size, void* d_ws, size_t ws_size,
                              hipStream_t stream) {
  (void)in_sizes; (void)n_in; (void)out_size; (void)ws_size;

  const float* x_a    = (const float*)d_in[0];
  const float* x_v    = (const float*)d_in[1];
  const float* x_t    = (const float*)d_in[2];
  const float* W_ih_a = (const float*)d_in[3];
  const float* W_hh_a = (const float*)d_in[4];
  const float* b_ih_a = (const float*)d_in[5];
  const float* b_hh_a = (const float*)d_in[6];
  const float* W_ih_v = (const float*)d_in[7];
  const float* W_hh_v = (const float*)d_in[8];
  const float* b_ih_v = (const float*)d_in[9];
  const float* b_hh_v = (const float*)d_in[10];
  const float* W_ih_t = (const float*)d_in[11];
  const float* W_hh_t = (const float*)d_in[12];
  const float* b_ih_t = (const float*)d_in[13];
  const float* b_hh_t = (const float*)d_in[14];
  const float* q_par  = (const float*)d_in[15];
  const float* W_r    = (const float*)d_in[16];
  const float* b_r    = (const float*)d_in[17];
  float* out = (float*)d_out;

  float* U_a = (float*)d_ws;                 // [MTOT, 16]
  float* U_v = U_a + (size_t)MTOT * HPAD;    // [MTOT, 16]
  float* U_t = U_v + (size_t)MTOT * HPAD;    // [MTOT, 16]
  float* x_p = U_t + (size_t)MTOT * HPAD;    // [BATCH, 30]

  const int gemm_blocks = MTOT / ROWS_PER_BLOCK;   // 400
  rnn_in_gemm< 74><<<gemm_blocks, 256, 0, stream>>>(x_a, W_ih_a, b_ih_a, b_hh_a, U_a);
  rnn_in_gemm< 35><<<gemm_blocks, 256, 0, stream>>>(x_v, W_ih_v, b_ih_v, b_hh_v, U_v);
  rnn_in_gemm<300><<<gemm_blocks, 256, 0, stream>>>(x_t, W_ih_t, b_ih_t, b_hh_t, U_t);

  rnn_recur<<<BATCH / 256, 256, 0, stream>>>(U_a, W_hh_a, x_p, 0);
  rnn_recur<<<BATCH / 256, 256, 0, stream>>>(U_v, W_hh_v, x_p, 10);
  rnn_recur<<<BATCH / 256, 256, 0, stream>>>(U_t, W_hh_t, x_p, 20);

  vqc<<<BATCH, 512, 0, stream>>>(x_p, q_par, W_r, b_r, out);
}